// CodirectEnhanceLayer_79499844649024
// MI455X (gfx1250) — compile-verified
//
#include <hip/hip_runtime.h>
#include <hip/hip_bf16.h>

#define N_NODES 100000
#define N_EDGES 1000000
#define DIM 64

typedef __attribute__((ext_vector_type(2))) float v2f;
typedef __attribute__((ext_vector_type(8))) float v8f;

// Workspace layout (floats):
//  [0..3]                : S0 (sum sqn[src]), S1 (sum sqn[dst]), inv_scale, pad
//  [4 .. 4+N)            : sqn[node]
//  [.. +N*64)            : src_diff  (stage-1 segment sum)
//  [.. +N*64)            : h_diff    (stage-2 segment sum)
//  [.. +E)               : gate[edge]
#define WS_SQN      4
#define WS_SRCDIFF  (WS_SQN + N_NODES)
#define WS_HDIFF    (WS_SRCDIFF + N_NODES * DIM)
#define WS_GATE     (WS_HDIFF + N_NODES * DIM)
#define WS_TOTAL    (WS_GATE + N_EDGES)

// ---------------------------------------------------------------- zero scratch
__global__ void k_zero(float* __restrict__ ws, long n) {
    long i = (long)blockIdx.x * blockDim.x + threadIdx.x;
    long stride = (long)gridDim.x * blockDim.x;
    for (; i < n; i += stride) ws[i] = 0.0f;
}

// ------------------------------------------------------ per-node squared norms
__global__ void k_sqnorm(const float* __restrict__ h, float* __restrict__ sqn) {
    int v = blockIdx.x * blockDim.x + threadIdx.x;
    if (v >= N_NODES) return;
    const float4* row = (const float4*)(h + v * DIM);
    float s = 0.f;
#pragma unroll
    for (int i = 0; i < DIM / 4; ++i) {
        float4 x = row[i];
        s += x.x * x.x + x.y * x.y + x.z * x.z + x.w * x.w;
    }
    sqn[v] = s;
}

// --------------------------------------- sum sqn over gathered src / dst edges
__global__ void k_norm_reduce(const int* __restrict__ src, const int* __restrict__ dst,
                              const float* __restrict__ sqn, float* __restrict__ S) {
    int i = blockIdx.x * blockDim.x + threadIdx.x;
    int stride = gridDim.x * blockDim.x;
    float ss = 0.f, sd = 0.f;
    for (; i < N_EDGES; i += stride) {
        ss += sqn[src[i]];
        sd += sqn[dst[i]];
    }
#pragma unroll
    for (int off = 16; off > 0; off >>= 1) {
        ss += __shfl_down(ss, off, 32);
        sd += __shfl_down(sd, off, 32);
    }
    if ((threadIdx.x & 31) == 0) {
        atomicAdd(&S[0], ss);
        atomicAdd(&S[1], sd);
    }
}

__global__ void k_finalize_scale(float* __restrict__ S) {
    S[2] = 1.0f / (sqrtf(S[0]) * sqrtf(S[1]) + 1e-6f);
}

// ---------------------------------------------------------------- edge kernel:
// per 16-edge tile (one wave32): gather h_src/h_dst once, fuse
//   (a) stage-1 scatter (h_src - h_dst) -> src_diff[dst]   (fp32 atomics, L2)
//   (b) P = h_src*h_dst*inv_scale  staged in LDS
//   (c) P[16x64] @ proj[64x64] via V_WMMA_F32_16X16X4_F32 (4 n-tiles x 16 k-tiles)
//   (d) relu, row-sum, clip, exp -> gate[e]
__global__ void __launch_bounds__(128)
k_edge_main(const float* __restrict__ h, const float* __restrict__ proj,
            const int* __restrict__ src, const int* __restrict__ dst,
            const float* __restrict__ S, float* __restrict__ src_diff,
            float* __restrict__ gate) {
    __shared__ float lds_proj[DIM * 65];      // padded rows: no bank conflicts
    __shared__ float lds_P[4][16 * 65];
    __shared__ float lds_Q[4][16 * 65];

    for (int idx = threadIdx.x; idx < DIM * DIM; idx += blockDim.x)
        lds_proj[(idx >> 6) * 65 + (idx & 63)] = proj[idx];
    __syncthreads();

    const int wave = threadIdx.x >> 5;
    const int lane = threadIdx.x & 31;
    const int l15  = lane & 15;
    const int half = lane >> 4;               // 0: K0/K1 lanes, 1: K2/K3 lanes
    float* P = lds_P[wave];
    float* Q = lds_Q[wave];

    const int base = (blockIdx.x * 4 + wave) * 16;   // first edge of this tile
    const float inv_scale = S[2];

    // lane e (and its mirror e+16) holds indices for edge e of the tile
    const int sidx = src[base + l15];
    const int didx = dst[base + l15];

    // gather + elementwise product + stage-1 scatter; half-wave per edge
#pragma unroll
    for (int t = 0; t < 8; ++t) {
        const int e  = 2 * t + half;                  // edge within tile
        const int su = __shfl(sidx, e, 32);
        const int du = __shfl(didx, e, 32);
        const float4 a = ((const float4*)(h + su * DIM))[l15];
        const float4 b = ((const float4*)(h + du * DIM))[l15];
        float* p = P + e * 65 + l15 * 4;
        p[0] = a.x * b.x * inv_scale;
        p[1] = a.y * b.y * inv_scale;
        p[2] = a.z * b.z * inv_scale;
        p[3] = a.w * b.w * inv_scale;
        float* sd = src_diff + du * DIM + l15 * 4;
        atomicAdd(sd + 0, a.x - b.x);
        atomicAdd(sd + 1, a.y - b.y);
        atomicAdd(sd + 2, a.z - b.z);
        atomicAdd(sd + 3, a.w - b.w);
    }

    // P[16x64] @ proj[64x64] -> relu -> Q[16x64]
#pragma unroll
    for (int nt = 0; nt < 4; ++nt) {
        v8f acc = {};
#pragma unroll
        for (int kt = 0; kt < 16; ++kt) {
            const int ka = kt * 4 + half * 2;         // K pair for this half-wave
            v2f a, b;
            a.x = P[l15 * 65 + ka];
            a.y = P[l15 * 65 + ka + 1];
            b.x = lds_proj[ka * 65 + nt * 16 + l15];
            b.y = lds_proj[(ka + 1) * 65 + nt * 16 + l15];
            acc = __builtin_amdgcn_wmma_f32_16x16x4_f32(
                false, a, false, b, (short)0, acc, false, false);
        }
#pragma unroll
        for (int i = 0; i < 8; ++i) {                 // C: vgpr i = row i (+8 for upper half)
            const int m = i + half * 8;
            float v = acc[i];
            Q[m * 65 + nt * 16 + l15] = v > 0.f ? v : 0.f;
        }
    }

    // gate[e] = exp(clip(row-sum, -5, 5)); per-wave DS ops are in-order
    if (lane < 16) {
        float s = 0.f;
#pragma unroll
        for (int j = 0; j < DIM; ++j) s += Q[lane * 65 + j];
        s = fminf(fmaxf(s, -5.0f), 5.0f);
        gate[base + lane] = __expf(s);
    }
}

// ------------------------- stage 2: h_diff[dst] += src_diff[src] * gate (L2 atomics)
__global__ void k_stage2(const int* __restrict__ src, const int* __restrict__ dst,
                         const float* __restrict__ src_diff, const float* __restrict__ gate,
                         float* __restrict__ h_diff) {
    const int tid = blockIdx.x * blockDim.x + threadIdx.x;   // E * 16 threads
    if (tid >= N_EDGES * 16) return;
    const int e = tid >> 4;
    const int c = tid & 15;
    const float g = gate[e];
    const int s = src[e], d = dst[e];
    const float4 v = ((const float4*)(src_diff + s * DIM))[c];
    float* o = h_diff + d * DIM + c * 4;
    atomicAdd(o + 0, v.x * g);
    atomicAdd(o + 1, v.y * g);
    atomicAdd(o + 2, v.z * g);
    atomicAdd(o + 3, v.w * g);
}

// --------------------------------- FFN: out = relu(h_diff @ W^T + b), WMMA f32
__global__ void __launch_bounds__(256)
k_ffn(const float* __restrict__ h_diff, const float* __restrict__ W,
      const float* __restrict__ bias, float* __restrict__ out) {
    const int wave = threadIdx.x >> 5;
    const int lane = threadIdx.x & 31;
    const int l15  = lane & 15;
    const int half = lane >> 4;
    const int base = (blockIdx.x * 8 + wave) * 16;   // 16-node tile (uniform per wave)
    if (base >= N_NODES) return;

    v2f afrag[16];
#pragma unroll
    for (int kt = 0; kt < 16; ++kt) {
        const float* p = h_diff + (base + l15) * DIM + kt * 4 + half * 2;
        afrag[kt].x = p[0];
        afrag[kt].y = p[1];
    }

#pragma unroll
    for (int nt = 0; nt < 4; ++nt) {
        const int col = nt * 16 + l15;
        v8f acc = {};
#pragma unroll
        for (int kt = 0; kt < 16; ++kt) {
            // B[k][col] = W[col][k]  (W^T falls out as a contiguous pair load)
            const float* q = W + col * DIM + kt * 4 + half * 2;
            v2f b;
            b.x = q[0];
            b.y = q[1];
            acc = __builtin_amdgcn_wmma_f32_16x16x4_f32(
                false, afrag[kt], false, b, (short)0, acc, false, false);
        }
        const float bb = bias[col];
#pragma unroll
        for (int i = 0; i < 8; ++i) {
            const int m = i + half * 8;
            const float v = acc[i] + bb;
            out[(base + m) * DIM + col] = v > 0.f ? v : 0.f;
        }
    }
}

// -----------------------------------------------------------------------------
extern "C" void kernel_launch(void* const* d_in, const int* in_sizes, int n_in,
                              void* d_out, int out_size, void* d_ws, size_t ws_size,
                              hipStream_t stream) {
    (void)in_sizes; (void)n_in; (void)out_size; (void)ws_size;
    const float* h    = (const float*)d_in[0];
    const float* proj = (const float*)d_in[1];
    const float* Wf   = (const float*)d_in[2];
    const float* bf   = (const float*)d_in[3];
    const int*   src  = (const int*)d_in[4];
    const int*   dst  = (const int*)d_in[5];

    float* ws       = (float*)d_ws;
    float* S        = ws;
    float* sqn      = ws + WS_SQN;
    float* src_diff = ws + WS_SRCDIFF;
    float* h_diff   = ws + WS_HDIFF;
    float* gate     = ws + WS_GATE;
    float* out      = (float*)d_out;

    // 1) zero accumulators (S, sqn, src_diff, h_diff)
    k_zero<<<4096, 256, 0, stream>>>(ws, (long)WS_GATE);
    // 2) per-node squared norms
    k_sqnorm<<<(N_NODES + 255) / 256, 256, 0, stream>>>(h, sqn);
    // 3) global norm reduction over edges
    k_norm_reduce<<<2048, 256, 0, stream>>>(src, dst, sqn, S);
    k_finalize_scale<<<1, 1, 0, stream>>>(S);
    // 4) fused edge pass: gate via WMMA + stage-1 scatter
    k_edge_main<<<N_EDGES / 64, 128, 0, stream>>>(h, proj, src, dst, S, src_diff, gate);
    // 5) stage-2 gated scatter
    k_stage2<<<(N_EDGES * 16) / 256, 256, 0, stream>>>(src, dst, src_diff, gate, h_diff);
    // 6) FFN via WMMA
    k_ffn<<<(N_NODES / 16 + 7) / 8, 256, 0, stream>>>(h_diff, Wf, bf, out);
}